// DeformableAttention2D_4037269258330
// MI455X (gfx1250) — compile-verified
//
#include <hip/hip_runtime.h>
#include <stdint.h>

// ---------------------------------------------------------------------------
// DeformableAttention2D forward for gfx1250 (MI455X), wave32 + WMMA f32.
// Sizes: B=4, N=256, DIM=256, HEADS=8, GROUPS=8, INNER=512, DIM_HEAD=64,
// H=W=4 (HW=16), CPB_HID=64, OFF_D=64.
// ---------------------------------------------------------------------------

typedef __attribute__((ext_vector_type(2))) float v2f;
typedef __attribute__((ext_vector_type(8))) float v8f;

// ---------------- wave-level 16x16 f32 WMMA tile --------------------------
// A element (m,k):  TA ? A[k*lda+m] : A[m*lda+k]
// B element (k,n):  TB ? B[n*ldb+k] : B[k*ldb+n]
// VGPR layouts per CDNA5 ISA 7.12.2:
//  A 16x4:  lanes 0-15 -> M=lane, vgpr0=K0,vgpr1=K1; lanes 16-31 -> K2,K3
//  B 4x16:  vgpr v: lanes0-15 -> K=v, N=lane; lanes16-31 -> K=v+2
//  C 16x16: vgpr v: lanes0-15 -> M=v; lanes16-31 -> M=v+8; N=lane&15
// K must be a multiple of 16. All branches are compile-time: the inner loop
// is 8 straight-line loads + 4 back-to-back v_wmma per 16-K chunk.
template <bool TA, bool TB>
__device__ __forceinline__ void wmma_tile16(const float* __restrict__ A, int lda,
                                            const float* __restrict__ B, int ldb,
                                            int K, v8f& acc) {
  const int lane = threadIdx.x & 31;
  const int r  = lane & 15;
  const int hi = lane >> 4;
  for (int k0 = 0; k0 < K; k0 += 16) {
    if (k0 + 16 < K) {  // prefetch next chunk (global_prefetch_b8)
      __builtin_prefetch(TA ? &A[(k0 + 16) * lda + r] : &A[r * lda + k0 + 16], 0, 1);
      __builtin_prefetch(TB ? &B[r * ldb + k0 + 16] : &B[(k0 + 16) * ldb + r], 0, 1);
    }
    v2f a[4], b[4];
#pragma unroll
    for (int s = 0; s < 4; ++s) {
      const int k = k0 + 4 * s + 2 * hi;
      if (TA) {
        a[s].x = A[k * lda + r];
        a[s].y = A[(k + 1) * lda + r];
      } else {
        a[s].x = A[r * lda + k];      // contiguous pair -> b64 load
        a[s].y = A[r * lda + k + 1];
      }
      if (TB) {
        b[s].x = B[r * ldb + k];      // contiguous pair -> b64 load
        b[s].y = B[r * ldb + k + 1];
      } else {
        b[s].x = B[k * ldb + r];
        b[s].y = B[(k + 1) * ldb + r];
      }
    }
#pragma unroll
    for (int s = 0; s < 4; ++s)
      acc = __builtin_amdgcn_wmma_f32_16x16x4_f32(false, a[s], false, b[s],
                                                  (short)0, acc, false, false);
  }
}

// ---------------- generic batched WMMA GEMM -------------------------------
// batch = blockIdx.y; operand offset = (batch/mod)*outer + (batch%mod)*inner
// BIAS: 0 = none, 1 = per output column (n), 2 = per output row (m)
template <bool TA, bool TB, bool TC, int BIAS>
__global__ __launch_bounds__(256)
void gemm_wmma(const float* __restrict__ A, const float* __restrict__ Bm,
               float* __restrict__ C, const float* __restrict__ bias,
               int M, int N, int K, int lda, int ldb, int ldc,
               long aOut, long aIn, int aMod,
               long bOut, long bIn, int bMod,
               long cOut, long cIn, int cMod,
               float alpha) {
  const int batch = blockIdx.y;
  const float* Ab = A + (long)(batch / aMod) * aOut + (long)(batch % aMod) * aIn;
  const float* Bb = Bm + (long)(batch / bMod) * bOut + (long)(batch % bMod) * bIn;
  float* Cb = C + (long)(batch / cMod) * cOut + (long)(batch % cMod) * cIn;

  const int wavesPerBlock = blockDim.x >> 5;
  const int wid = blockIdx.x * wavesPerBlock + (threadIdx.x >> 5);
  const int tilesN = N >> 4, tilesM = M >> 4;
  if (wid >= tilesM * tilesN) return;   // wave-uniform: EXEC stays all-1s
  const int tm = wid / tilesN, tn = wid % tilesN;
  const int m0 = tm << 4, n0 = tn << 4;

  const float* At = TA ? (Ab + m0) : (Ab + m0 * lda);
  const float* Bt = TB ? (Bb + n0 * ldb) : (Bb + n0);

  v8f acc = {0.f, 0.f, 0.f, 0.f, 0.f, 0.f, 0.f, 0.f};
  wmma_tile16<TA, TB>(At, lda, Bt, ldb, K, acc);

  const int lane = threadIdx.x & 31;
  const int r = lane & 15, hi = lane >> 4;
#pragma unroll
  for (int v = 0; v < 8; ++v) {
    const int m = m0 + v + 8 * hi;
    const int n = n0 + r;
    float val = alpha * acc[v];
    if (BIAS == 1) val += bias[n];
    if (BIAS == 2) val += bias[m];
    if (TC) Cb[n * ldc + m] = val;
    else    Cb[m * ldc + n] = val;
  }
}

// ---------------- elementwise / prep kernels ------------------------------

// qin[b,n,d] = pose_feat[b,d,n] + point_emb(pose_init)[b,n,d]
__global__ void qin_kernel(const float* __restrict__ pose_feat,
                           const float* __restrict__ pose_init,
                           const float* __restrict__ pe_gauss,
                           float* __restrict__ qin) {
  int idx = blockIdx.x * blockDim.x + threadIdx.x;
  if (idx >= 4 * 256 * 256) return;
  int d = idx & 255, n = (idx >> 8) & 255, b = idx >> 16;
  float px = pose_init[b * 512 + n];
  float py = pose_init[b * 512 + 256 + n];
  int j = d & 127;
  float c = ((2.f * px - 1.f) * pe_gauss[j] +
             (2.f * py - 1.f) * pe_gauss[128 + j]) * 6.283185307179586f;
  float emb = (d < 128) ? sinf(c) : cosf(c);
  qin[(b * 256 + n) * 256 + d] = pose_feat[(b * 256 + d) * 256 + n] + emb;
}

// kvin[b,t,d] = rgb_feat[b,d,t] + sinusoid(t,d)
__global__ void kvin_kernel(const float* __restrict__ rgb, float* __restrict__ kvin) {
  int idx = blockIdx.x * blockDim.x + threadIdx.x;
  if (idx >= 4 * 16 * 256) return;
  int d = idx & 255, t = (idx >> 8) & 15, b = idx >> 12;
  float e = (float)(d & ~1) / 256.f;
  float ang = (float)t * powf(10000.f, -e);
  float pe = ((d & 1) == 0) ? sinf(ang) : cosf(ang);
  kvin[(b * 16 + t) * 256 + d] = rgb[(b * 256 + d) * 16 + t] + pe;
}

// cross-attention core: per (b,h,i) row, 16 kv tokens, head dim 32
__global__ void xattn_kernel(const float* __restrict__ qx, const float* __restrict__ kx,
                             const float* __restrict__ vx, float* __restrict__ xo) {
  int t = blockIdx.x * blockDim.x + threadIdx.x;
  if (t >= 4 * 8 * 256) return;
  int b = t >> 11, h = (t >> 8) & 7, i = t & 255;
  const float scale = 0.1767766952966369f;  // 1/sqrt(32)
  float qv[32];
  const float* qb = qx + (b * 256 + i) * 256 + h * 32;
#pragma unroll
  for (int d = 0; d < 32; ++d) qv[d] = qb[d];
  float s[16], mx = -3.4e38f;
  for (int j = 0; j < 16; ++j) {
    const float* kb = kx + (b * 16 + j) * 256 + h * 32;
    float acc = 0.f;
#pragma unroll
    for (int d = 0; d < 32; ++d) acc += qv[d] * kb[d];
    s[j] = acc * scale;
    mx = fmaxf(mx, s[j]);
  }
  float sum = 0.f;
  for (int j = 0; j < 16; ++j) { s[j] = expf(s[j] - mx); sum += s[j]; }
  float inv = 1.f / sum;
  float* ob = xo + (b * 256 + i) * 256 + h * 32;
#pragma unroll
  for (int d = 0; d < 32; ++d) {
    float acc = 0.f;
    for (int j = 0; j < 16; ++j)
      acc += s[j] * vx[(b * 16 + j) * 256 + h * 32 + d];
    ob[d] = acc * inv;
  }
}

// xsum[b,d,n] = pose_feat[b,d,n] + pose_cross[b,n,d]
__global__ void xsum_kernel(const float* __restrict__ pose_feat,
                            const float* __restrict__ pcross,
                            float* __restrict__ xsum) {
  int idx = blockIdx.x * blockDim.x + threadIdx.x;
  if (idx >= 4 * 256 * 256) return;
  int n = idx & 255, d = (idx >> 8) & 255, b = idx >> 16;
  xsum[(b * 256 + d) * 256 + n] =
      pose_feat[(b * 256 + d) * 256 + n] + pcross[(b * 256 + n) * 256 + d];
}

// offsets: depthwise GELU + 64->2 conv, tanh scale; emit grid and vgrid (x,y)
__global__ void offsets_kernel(const float* __restrict__ qbuf,
                               const float* __restrict__ pose_init,
                               const float* __restrict__ w1, const float* __restrict__ b1,
                               const float* __restrict__ w2,
                               float* __restrict__ gridc, float* __restrict__ vgridc) {
  int idx = blockIdx.x * blockDim.x + threadIdx.x;
  if (idx >= 32 * 256) return;
  int n = idx & 255, bg = idx >> 8, b = bg >> 3, g = bg & 7;
  float a0 = 0.f, a1 = 0.f;
  for (int c = 0; c < 64; ++c) {
    float gq = qbuf[(b * 512 + g * 64 + c) * 256 + n];
    float u = gq * w1[c] + b1[c];
    float o = 0.5f * u * (1.f + erff(u * 0.7071067811865476f));  // exact GELU
    a0 += w2[c] * o;
    a1 += w2[64 + c] * o;
  }
  float off0 = tanhf(a0) * (4.f / 6.f);
  float off1 = tanhf(a1) * (4.f / 6.f);
  float gx = pose_init[b * 512 + n] * 2.f - 1.f;
  float gy = pose_init[b * 512 + 256 + n] * 2.f - 1.f;
  gridc[bg * 512 + 2 * n] = gx;
  gridc[bg * 512 + 2 * n + 1] = gy;
  vgridc[bg * 512 + 2 * n] = gx + off0;
  vgridc[bg * 512 + 2 * n + 1] = gy + off1;
}

// bilinear grid-sample of rgb (4x4, zero pad, align_corners=False)
__global__ void gridsample_kernel(const float* __restrict__ rgb,
                                  const float* __restrict__ vgridc,
                                  float* __restrict__ kvfeat) {
  int idx = blockIdx.x * blockDim.x + threadIdx.x;
  if (idx >= 32 * 32 * 256) return;
  int n = idx & 255, c = (idx >> 8) & 31, bg = idx >> 13;
  int b = bg >> 3, g = bg & 7, ch = g * 32 + c;
  float vx = vgridc[bg * 512 + 2 * n];
  float vy = vgridc[bg * 512 + 2 * n + 1];
  float x = (vx + 1.f) * 2.f - 0.5f;
  float y = (vy + 1.f) * 2.f - 0.5f;
  float x0f = floorf(x), y0f = floorf(y);
  float wx = x - x0f, wy = y - y0f;
  int x0 = (int)x0f, y0 = (int)y0f;
  const float* img = rgb + (b * 256 + ch) * 16;
  auto samp = [&](int yi, int xi) -> float {
    bool ok = (xi >= 0) && (xi < 4) && (yi >= 0) && (yi < 4);
    int xc = xi < 0 ? 0 : (xi > 3 ? 3 : xi);
    int yc = yi < 0 ? 0 : (yi > 3 ? 3 : yi);
    return ok ? img[yc * 4 + xc] : 0.f;
  };
  float v00 = samp(y0, x0),     v01 = samp(y0, x0 + 1);
  float v10 = samp(y0 + 1, x0), v11 = samp(y0 + 1, x0 + 1);
  float out = v00 * (1.f - wx) * (1.f - wy) + v01 * wx * (1.f - wy) +
              v10 * (1.f - wx) * wy + v11 * wx * wy;
  kvfeat[(b * 256 + ch) * 256 + n] = out;
}

// ---------------- fused CPB MLP (dominant compute, WMMA + LDS) ------------
// Each block handles 128 (bg,i,j) points: W1 staged into LDS via async DMA
// (GLOBAL_LOAD_ASYNC_TO_LDS_B128, ASYNCcnt) overlapping layer0 VALU work;
// layer0 (2->64) in VALU to LDS; layer1 (64->64) as 8x4 WMMA tiles with
// A-fragments hoisted into registers; layer2 (64->1) folded into the
// epilogue via LDS float atomics.
__global__ __launch_bounds__(256)
void cpb_kernel(const float* __restrict__ gridc, const float* __restrict__ vgridc,
                const float* __restrict__ w0, const float* __restrict__ b0,
                const float* __restrict__ w1, const float* __restrict__ b1,
                const float* __restrict__ w2, const float* __restrict__ b2,
                float* __restrict__ biasb) {
  __shared__ float sH0[128 * 64];
  __shared__ __align__(16) float sW1[64 * 64];
  __shared__ float sS[128 * 2];
  __shared__ float sPsum[128];
  const int tid = threadIdx.x;
  const long pbase = (long)blockIdx.x * 128;

  // async DMA of W1 (16KB) into LDS: 4 x b128 per thread, tracked by ASYNCcnt
#pragma unroll
  for (int t4 = 0; t4 < 4; ++t4) {
    int idx = tid * 4 + t4 * 1024;
    unsigned lds_addr = (unsigned)(uintptr_t)(&sW1[idx]);  // low 32b = LDS offset
    unsigned long long gaddr = (unsigned long long)(uintptr_t)(w1 + idx);
    asm volatile("global_load_async_to_lds_b128 %0, %1, off"
                 :: "v"(lds_addr), "v"(gaddr) : "memory");
  }

  if (tid < 128) {
    long p = pbase + tid;
    int j = (int)(p & 255);
    int i = (int)((p >> 8) & 255);
    int bg = (int)(p >> 16);
    float d0 = gridc[bg * 512 + 2 * i]     - vgridc[bg * 512 + 2 * j];
    float d1 = gridc[bg * 512 + 2 * i + 1] - vgridc[bg * 512 + 2 * j + 1];
    sS[tid * 2]     = (d0 >= 0.f) ? log1pf(d0) : -log1pf(-d0);
    sS[tid * 2 + 1] = (d1 >= 0.f) ? log1pf(d1) : -log1pf(-d1);
    sPsum[tid] = 0.f;
  }
  __syncthreads();

  for (int idx = tid; idx < 128 * 64; idx += 256) {
    int pl = idx >> 6, h = idx & 63;
    float v = w0[h * 2] * sS[pl * 2] + w0[h * 2 + 1] * sS[pl * 2 + 1] + b0[h];
    sH0[pl * 64 + h] = fmaxf(v, 0.f);
  }
  // all waves drain their own async copies, then barrier -> sW1/sH0 visible
  asm volatile("s_wait_asynccnt 0x0" ::: "memory");
  __syncthreads();

  const int wave = tid >> 5, lane = tid & 31;
  const int r = lane & 15, hi = lane >> 4;
  const float* At = &sH0[wave * 16 * 64];  // this wave's 16 points

  v2f afrag[16];  // hoisted A panel: read once, reused for all 4 B tiles
#pragma unroll
  for (int s = 0; s < 16; ++s) {
    const int k = 4 * s + 2 * hi;
    afrag[s].x = At[r * 64 + k];
    afrag[s].y = At[r * 64 + k + 1];
  }

#pragma unroll
  for (int ct = 0; ct < 4; ++ct) {
    v8f acc = {0.f, 0.f, 0.f, 0.f, 0.f, 0.f, 0.f, 0.f};
    const float* Bt = &sW1[ct * 16 * 64];  // W1 rows (output units) ct*16..+15
#pragma unroll
    for (int s = 0; s < 16; ++s) {
      const int k = 4 * s + 2 * hi;
      v2f bf;
      bf.x = Bt[r * 64 + k];      // B(k,n)=W1[n][k] -> contiguous b64
      bf.y = Bt[r * 64 + k + 1];
      acc = __builtin_amdgcn_wmma_f32_16x16x4_f32(false, afrag[s], false, bf,
                                                  (short)0, acc, false, false);
    }
#pragma unroll
    for (int v = 0; v < 8; ++v) {
      int m = wave * 16 + v + 8 * hi;  // local point
      int n = ct * 16 + r;             // hidden unit
      float h1 = fmaxf(acc[v] + b1[n], 0.f);
      atomicAdd(&sPsum[m], h1 * w2[n]);
    }
  }
  __syncthreads();
  if (tid < 128) biasb[pbase + tid] = sPsum[tid] + b2[0];
}

// softmax over rows of sim + bias (row = (b,h,i), 256 cols)
__global__ __launch_bounds__(256)
void softmax_kernel(float* __restrict__ sim, const float* __restrict__ biasb) {
  __shared__ float red[256];
  const long off = (long)blockIdx.x * 256;
  const int t = threadIdx.x;
  float v = sim[off + t] + biasb[off + t];
  red[t] = v;
  __syncthreads();
  for (int s = 128; s > 0; s >>= 1) {
    if (t < s) red[t] = fmaxf(red[t], red[t + s]);
    __syncthreads();
  }
  float mx = red[0];
  __syncthreads();
  float e = expf(v - mx);
  red[t] = e;
  __syncthreads();
  for (int s = 128; s > 0; s >>= 1) {
    if (t < s) red[t] += red[t + s];
    __syncthreads();
  }
  sim[off + t] = e / red[0];
}

// ---------------- host-side launch ----------------------------------------

template <bool TA, bool TB, bool TC, int BIAS>
static void launchGemm(hipStream_t s, const float* A, const float* Bm, float* C,
                       const float* bias, int M, int N, int K,
                       int lda, int ldb, int ldc,
                       long aOut, long aIn, int aMod,
                       long bOut, long bIn, int bMod,
                       long cOut, long cIn, int cMod,
                       float alpha, int batches) {
  int tiles = (M / 16) * (N / 16);
  dim3 grid((tiles + 7) / 8, batches);
  gemm_wmma<TA, TB, TC, BIAS><<<grid, 256, 0, s>>>(
      A, Bm, C, bias, M, N, K, lda, ldb, ldc,
      aOut, aIn, aMod, bOut, bIn, bMod, cOut, cIn, cMod, alpha);
}

extern "C" void kernel_launch(void* const* d_in, const int* in_sizes, int n_in,
                              void* d_out, int out_size, void* d_ws, size_t ws_size,
                              hipStream_t stream) {
  const float* pose_feat = (const float*)d_in[0];
  const float* rgb_feat  = (const float*)d_in[1];
  const float* pose_init = (const float*)d_in[2];
  const float* mha_in_w  = (const float*)d_in[3];
  const float* mha_in_b  = (const float*)d_in[4];
  const float* mha_out_w = (const float*)d_in[5];
  const float* mha_out_b = (const float*)d_in[6];
  const float* pe_gauss  = (const float*)d_in[7];
  const float* off_w1    = (const float*)d_in[8];
  const float* off_b1    = (const float*)d_in[9];
  const float* off_w2    = (const float*)d_in[10];
  const float* cpb_w0    = (const float*)d_in[11];
  const float* cpb_b0    = (const float*)d_in[12];
  const float* cpb_w1    = (const float*)d_in[13];
  const float* cpb_b1    = (const float*)d_in[14];
  const float* cpb_w2    = (const float*)d_in[15];
  const float* cpb_b2    = (const float*)d_in[16];
  const float* q_w       = (const float*)d_in[17];
  const float* k_w       = (const float*)d_in[18];
  const float* v_w       = (const float*)d_in[19];
  const float* out_w     = (const float*)d_in[20];
  const float* out_b     = (const float*)d_in[21];
  float* out = (float*)d_out;

  // ---- workspace layout (floats) ----
  float* ws = (float*)d_ws;
  float* qin     = ws;             // 262144
  float* kvin    = qin + 262144;   // 16384
  float* qx      = kvin + 16384;   // 262144
  float* kx      = qx + 262144;    // 16384
  float* vx      = kx + 16384;     // 16384
  float* xattn   = vx + 16384;     // 262144
  float* pcross  = xattn + 262144; // 262144
  float* xsum    = pcross + 262144;// 262144
  float* qbuf    = xsum + 262144;  // 524288
  float* gridc   = qbuf + 524288;  // 16384
  float* vgridc  = gridc + 16384;  // 16384
  float* kvfeat  = vgridc + 16384; // 262144
  float* kbuf    = kvfeat + 262144;// 524288
  float* vbuf    = kbuf + 524288;  // 524288
  float* simbuf  = vbuf + 524288;  // 2097152
  float* biasb   = simbuf + 2097152; // 2097152
  float* attnout = biasb + 2097152;  // 524288

  // 1) query input = pose^T + random-Fourier point embedding
  qin_kernel<<<1024, 256, 0, stream>>>(pose_feat, pose_init, pe_gauss, qin);
  // 2) kv input = rgb tokens + sinusoid table
  kvin_kernel<<<64, 256, 0, stream>>>(rgb_feat, kvin);
  // 3-5) MHA in-projections (X @ W^T + b), W rows [0:256),[256:512),[512:768)
  launchGemm<false, true, false, 1>(stream, qin, mha_in_w, qx, mha_in_b,
      256, 256, 256, 256, 256, 256,
      65536, 0, 1, 0, 0, 1, 65536, 0, 1, 1.f, 4);
  launchGemm<false, true, false, 1>(stream, kvin, mha_in_w + 65536, kx, mha_in_b + 256,
      16, 256, 256, 256, 256, 256,
      4096, 0, 1, 0, 0, 1, 4096, 0, 1, 1.f, 4);
  launchGemm<false, true, false, 1>(stream, kvin, mha_in_w + 131072, vx, mha_in_b + 512,
      16, 256, 256, 256, 256, 256,
      4096, 0, 1, 0, 0, 1, 4096, 0, 1, 1.f, 4);
  // 6) cross-attention (8 heads x 16 kv tokens)
  xattn_kernel<<<32, 256, 0, stream>>>(qx, kx, vx, xattn);
  // 7) MHA out-projection
  launchGemm<false, true, false, 1>(stream, xattn, mha_out_w, pcross, mha_out_b,
      256, 256, 256, 256, 256, 256,
      65536, 0, 1, 0, 0, 1, 65536, 0, 1, 1.f, 4);
  // 8) xsum = pose_feat + pose_cross^T
  xsum_kernel<<<1024, 256, 0, stream>>>(pose_feat, pcross, xsum);
  // 9) grouped 1x1: q = W_g @ x_g   (batch = b*8+g)
  launchGemm<false, false, false, 0>(stream, q_w, xsum, qbuf, nullptr,
      64, 256, 32, 32, 256, 256,
      0, 2048, 8, 65536, 8192, 8, 131072, 16384, 8, 1.f, 32);
  // 10) offsets + grids
  offsets_kernel<<<32, 256, 0, stream>>>(qbuf, pose_init, off_w1, off_b1, off_w2,
                                         gridc, vgridc);
  // 11) bilinear grid sample
  gridsample_kernel<<<1024, 256, 0, stream>>>(rgb_feat, vgridc, kvfeat);
  // 12-13) grouped 1x1 for k, v on sampled features
  launchGemm<false, false, false, 0>(stream, k_w, kvfeat, kbuf, nullptr,
      64, 256, 32, 32, 256, 256,
      0, 2048, 8, 65536, 8192, 8, 131072, 16384, 8, 1.f, 32);
  launchGemm<false, false, false, 0>(stream, v_w, kvfeat, vbuf, nullptr,
      64, 256, 32, 32, 256, 256,
      0, 2048, 8, 65536, 8192, 8, 131072, 16384, 8, 1.f, 32);
  // 14) sim = (q * 64^-0.5)^T @ k  per (b,h): transA on q (stored [d,i])
  launchGemm<true, false, false, 0>(stream, qbuf, kbuf, simbuf, nullptr,
      256, 256, 64, 256, 256, 256,
      16384, 0, 1, 16384, 0, 1, 65536, 0, 1, 0.125f, 32);
  // 15) CPB bias MLP (fused, WMMA + LDS + async DMA) over 2.1M points
  cpb_kernel<<<16384, 256, 0, stream>>>(gridc, vgridc, cpb_w0, cpb_b0, cpb_w1,
                                        cpb_b1, cpb_w2, cpb_b2, biasb);
  // 16) softmax(sim + bias) in place
  softmax_kernel<<<8192, 256, 0, stream>>>(simbuf, biasb);
  // 17) out_h = attn @ v^T  (B stored [d,j] -> transB; store transposed [d,i])
  launchGemm<false, true, true, 0>(stream, simbuf, vbuf, attnout, nullptr,
      256, 64, 256, 256, 256, 256,
      65536, 0, 1, 16384, 0, 1, 16384, 0, 1, 1.f, 32);
  // 18) final projection: out = out_w @ attnout + out_b (bias per output row)
  launchGemm<false, false, false, 2>(stream, out_w, attnout, out, out_b,
      256, 256, 512, 512, 256, 256,
      0, 0, 1, 131072, 0, 1, 65536, 0, 1, 1.f, 4);
}